// SyntheticEMGInjection_35321811042875
// MI455X (gfx1250) — compile-verified
//
#include <hip/hip_runtime.h>
#include <math.h>

// ---------------------------------------------------------------------------
// SyntheticEMGInjection for MI455X (gfx1250) — fused single-pass version.
// One 256-thread block per (b,c) row. The whole noise row (+halo) is staged
// into LDS with CDNA5 async global->LDS b128 bursts (ASYNCcnt path); the
// sum(x^2) phase runs concurrently with the in-flight async burst. HBM
// traffic: read x once, read noise once, write out once (~756 MB @ 23.3 TB/s
// => ~32 us roofline).
// ---------------------------------------------------------------------------

#define B_N     64
#define C_N     64
#define T_LEN   15360
#define KW      25
#define HALO    12               // KW/2
#define TPB     256
#define TILE    1024             // 4 elements per thread per tile
#define TILES   (T_LEN / TILE)   // 15, exact
#define SH_N    (T_LEN + 2 * HALO)   // 15384 floats = 61536 B

#define PROB    0.4f
#define SNR_LO  0.5f
#define SNR_HI  3.0f
#define BF_LO   0.1f
#define BF_HI   0.5f
#define MAX_CH  0.5f

// ---- CDNA5 async global->LDS support probes -------------------------------
#if defined(__has_builtin)
#  if __has_builtin(__builtin_amdgcn_global_load_async_to_lds_b128)
#    define ASYNC_MODE 2
#  elif __has_builtin(__builtin_amdgcn_global_load_async_to_lds_b32)
#    define ASYNC_MODE 1
#  endif
#  if __has_builtin(__builtin_amdgcn_s_wait_asynccnt)
#    define HAVE_WAIT_ASYNC_BUILTIN 1
#  endif
#endif
#ifndef ASYNC_MODE
#  define ASYNC_MODE 0
#endif

typedef int v4i __attribute__((vector_size(4 * sizeof(int))));
typedef __attribute__((address_space(1))) int g_as_int;   // global, b32
typedef __attribute__((address_space(3))) int l_as_int;   // LDS, b32
typedef __attribute__((address_space(1))) v4i g_as_v4i;   // global, b128
typedef __attribute__((address_space(3))) v4i l_as_v4i;   // LDS, b128

__device__ __forceinline__ void wait_async_zero() {
#if ASYNC_MODE > 0
#  if defined(HAVE_WAIT_ASYNC_BUILTIN)
  __builtin_amdgcn_s_wait_asynccnt(0);
#  else
  asm volatile("s_wait_asynccnt 0" ::: "memory");
#  endif
#endif
}

// Per-batch burst-mask parameters (replicates the JAX f32/int32 math exactly).
struct MaskParams {
  float sf, endf, ef, denom;
};

__device__ __forceinline__ MaskParams mask_params(float bu, float su) {
  int blen = (int)floorf((float)T_LEN * (BF_LO + bu * (BF_HI - BF_LO)));
  int tmb = T_LEN - blen; if (tmb < 1) tmb = 1;
  int start = (int)floorf(su * (float)tmb);
  int elen = blen / 4; if (elen > 25) elen = 25;
  MaskParams p;
  p.sf = (float)start;
  p.endf = (float)start + (float)blen;
  p.ef = (float)elen;
  p.denom = fmaxf(p.ef - 1.0f, 1.0f);
  return p;
}

__device__ __forceinline__ float burst_mask(const MaskParams& p, float tt) {
  float up = (tt - p.sf) / p.denom;
  float down = (p.endf - 1.0f - tt) / p.denom;
  float ramped = (tt < p.sf + p.ef) ? up : ((tt >= p.endf - p.ef) ? down : 1.0f);
  float inb = (tt >= p.sf && tt < p.endf) ? 1.0f : 0.0f;
  float core = (p.ef > 1.0f) ? ramped : 1.0f;
  return inb * core;
}

// ---------------------------------------------------------------------------
// Fused kernel: one block per (b,c) row.
// ---------------------------------------------------------------------------
__global__ void __launch_bounds__(TPB)
emg_fused_kernel(const float* __restrict__ x, const float* __restrict__ noise,
                 const float* __restrict__ apply_u, const float* __restrict__ frac_u,
                 const float* __restrict__ ch_u, const float* __restrict__ burst_u,
                 const float* __restrict__ start_u, const float* __restrict__ snr_u,
                 float* __restrict__ out) {
  __shared__ float shn[SH_N];    // zero-padded noise row: element t -> shn[t+HALO]
  __shared__ float red[TPB];
  __shared__ float gsh;

  const int r = blockIdx.x;      // row index in [0, B*C)
  const int b = r / C_N;
  const int c = r - b * C_N;
  const int tid = threadIdx.x;

  const float* nrow = noise + (size_t)r * T_LEN;
  const float* xrow = x + (size_t)r * T_LEN;
  float* orow = out + (size_t)r * T_LEN;

  const MaskParams mp = mask_params(burst_u[b], start_u[b]);

  // ---- 1) zero the halo (DS path) ----
  if (tid < HALO) {
    shn[tid] = 0.0f;
    shn[HALO + T_LEN + tid] = 0.0f;
  }

  // ---- 2) async-stage the whole noise row into LDS (b128 bursts) ----
#pragma unroll
  for (int tile = 0; tile < TILES; ++tile) {
    const int idx = tile * TILE + (tid << 2);          // 4 floats / thread / tile
#if ASYNC_MODE == 2
    __builtin_amdgcn_global_load_async_to_lds_b128(
        (g_as_v4i*)(nrow + idx), (l_as_v4i*)(shn + HALO + idx), 0, 0);
#elif ASYNC_MODE == 1
    __builtin_amdgcn_global_load_async_to_lds_b32(
        (g_as_int*)(nrow + idx + 0), (l_as_int*)(shn + HALO + idx + 0), 0, 0);
    __builtin_amdgcn_global_load_async_to_lds_b32(
        (g_as_int*)(nrow + idx + 1), (l_as_int*)(shn + HALO + idx + 1), 0, 0);
    __builtin_amdgcn_global_load_async_to_lds_b32(
        (g_as_int*)(nrow + idx + 2), (l_as_int*)(shn + HALO + idx + 2), 0, 0);
    __builtin_amdgcn_global_load_async_to_lds_b32(
        (g_as_int*)(nrow + idx + 3), (l_as_int*)(shn + HALO + idx + 3), 0, 0);
#else
    const float4 nv = *reinterpret_cast<const float4*>(nrow + idx);
    *reinterpret_cast<float4*>(shn + HALO + idx) = nv;
#endif
  }

  // ---- 3) sum(x^2) from global; overlaps with the in-flight async burst ----
  float accx = 0.0f;
  for (int tile = 0; tile < TILES; ++tile) {
    const int idx = tile * TILE + (tid << 2);
    const float4 xv = *reinterpret_cast<const float4*>(xrow + idx);
    accx = fmaf(xv.x, xv.x, accx);
    accx = fmaf(xv.y, xv.y, accx);
    accx = fmaf(xv.z, xv.z, accx);
    accx = fmaf(xv.w, xv.w, accx);
  }

  // ---- 4) drain ASYNCcnt, make LDS visible to all waves ----
  wait_async_zero();
  __syncthreads();

  // ---- 5) sum((emg*mask)^2) via incremental 25-tap sliding window ----
  float acce = 0.0f;
  for (int tile = 0; tile < TILES; ++tile) {
    const int t0 = tile * TILE + (tid << 2);
    float w = 0.0f;
#pragma unroll
    for (int j = 0; j < KW; ++j) w += shn[t0 + j];
#pragma unroll
    for (int e = 0; e < 4; ++e) {
      float emg = shn[t0 + e + HALO] - w * (1.0f / (float)KW);
      float m = burst_mask(mp, (float)(t0 + e));
      float ev = emg * m;
      acce = fmaf(ev, ev, acce);
      if (e < 3) w = w - shn[t0 + e] + shn[t0 + e + KW];
    }
  }

  // ---- 6) block reductions ----
  red[tid] = accx;
  __syncthreads();
  for (int off = TPB / 2; off > 0; off >>= 1) {
    if (tid < off) red[tid] += red[tid + off];
    __syncthreads();
  }
  const float sumx = red[0];
  __syncthreads();

  red[tid] = acce;
  __syncthreads();
  for (int off = TPB / 2; off > 0; off >>= 1) {
    if (tid < off) red[tid] += red[tid + off];
    __syncthreads();
  }

  // ---- 7) per-row gain: sel * sig_p / (noi_p * snr) ----
  if (tid == 0) {
    const float sume = red[0];
    float sig_p = sqrtf(sumx / (float)T_LEN) + 1e-8f;
    float noi_p = sqrtf(sume / (float)T_LEN) + 1e-8f;
    float snr = SNR_LO + snr_u[b] * (SNR_HI - SNR_LO);

    // stable rank == argsort(argsort(ch_u)) semantics (ties -> lower index first)
    float myv = ch_u[b * C_N + c];
    int rank = 0;
    for (int j = 0; j < C_N; ++j) {
      float v = ch_u[b * C_N + j];
      rank += (v < myv) || (v == myv && j < c);
    }
    int naff = (int)floorf((float)C_N * frac_u[b] * MAX_CH);
    if (naff < 1) naff = 1;
    float sel = (apply_u[b] <= PROB && rank < naff) ? 1.0f : 0.0f;
    gsh = sel * sig_p / (noi_p * snr);
  }
  __syncthreads();
  const float gv = gsh;

  // ---- 8) output: out = x + g * emg * mask (x re-read is L2-hot) ----
  for (int tile = 0; tile < TILES; ++tile) {
    const int t0 = tile * TILE + (tid << 2);
    const float4 xv = *reinterpret_cast<const float4*>(xrow + t0);
    const float xin[4] = {xv.x, xv.y, xv.z, xv.w};

    float w = 0.0f;
#pragma unroll
    for (int j = 0; j < KW; ++j) w += shn[t0 + j];

    float res[4];
#pragma unroll
    for (int e = 0; e < 4; ++e) {
      float emg = shn[t0 + e + HALO] - w * (1.0f / (float)KW);
      float m = burst_mask(mp, (float)(t0 + e));
      res[e] = fmaf(gv, emg * m, xin[e]);
      if (e < 3) w = w - shn[t0 + e] + shn[t0 + e + KW];
    }

    *reinterpret_cast<float4*>(orow + t0) =
        make_float4(res[0], res[1], res[2], res[3]);
  }
}

// ---------------------------------------------------------------------------
extern "C" void kernel_launch(void* const* d_in, const int* in_sizes, int n_in,
                              void* d_out, int out_size, void* d_ws, size_t ws_size,
                              hipStream_t stream) {
  (void)in_sizes; (void)n_in; (void)out_size; (void)d_ws; (void)ws_size;

  const float* x       = (const float*)d_in[0];
  const float* noise   = (const float*)d_in[1];
  const float* apply_u = (const float*)d_in[2];
  const float* frac_u  = (const float*)d_in[3];
  const float* ch_u    = (const float*)d_in[4];
  const float* burst_u = (const float*)d_in[5];
  const float* start_u = (const float*)d_in[6];
  const float* snr_u   = (const float*)d_in[7];
  float* out = (float*)d_out;

  emg_fused_kernel<<<B_N * C_N, TPB, 0, stream>>>(
      x, noise, apply_u, frac_u, ch_u, burst_u, start_u, snr_u, out);
}